// InverseRadonTransform_76596446757205
// MI455X (gfx1250) — compile-verified
//
#include <hip/hip_runtime.h>
#include <math.h>

#define DET   512
#define NANG  512
#define OUT   362
#define NCOL  1024          // B * NANG columns
#define PI_F  3.14159265358979323846f

typedef __attribute__((ext_vector_type(2))) float v2f;
typedef __attribute__((ext_vector_type(4))) float v4f;
typedef __attribute__((ext_vector_type(8))) float v8f;

// ---------------------------------------------------------------------------
// Setup: build the Toeplitz ramp-filter matrix H[n][k] = ramp(|n-k|) and the
// per-angle sin/cos tables. (FFT of the reference filter reduces exactly to
// circular convolution with h = 2*f since f is circularly symmetric.)
// ---------------------------------------------------------------------------
__global__ void __launch_bounds__(256)
fbp_setup(float* __restrict__ H, float* __restrict__ cosT, float* __restrict__ sinT) {
    const int idx = blockIdx.x * blockDim.x + threadIdx.x;
    if (idx < DET * DET) {
        const int n = idx >> 9;
        const int k = idx & (DET - 1);
        int d = n - k; if (d < 0) d = -d;
        float val = 0.0f;
        if (d == 0)        val = 0.5f;                                   // 2 * 0.25
        else if (d & 1)    val = -2.0f / (PI_F * PI_F * (float)d * (float)d);
        H[idx] = val;
    }
    if (idx < NANG) {
        // theta = deg2rad(idx * 180/512) = idx * pi/512
        const float th = (float)idx * (PI_F / (float)NANG);
        float s, c;
        __sincosf(th, &s, &c);
        cosT[idx] = c;
        sinT[idx] = s;
    }
}

// ---------------------------------------------------------------------------
// Ramp filtering as FP32 WMMA GEMM:  F[n,c] = sum_k H[n,k] * S[k,c]
//   M = n (512), N = c (1024), K = k (512), V_WMMA_F32_16X16X4_F32.
// One wave per 16x16 output tile; 2048 waves = 256 blocks * 8 waves.
// Output stored transposed: Ft[c*512 + n]  (detector contiguous per angle).
// ---------------------------------------------------------------------------
__global__ void __launch_bounds__(256)
fbp_filter_wmma(const float* __restrict__ sino,   // [2][DET][NANG]
                const float* __restrict__ H,      // [DET][DET]
                float* __restrict__ Ft) {         // [NCOL][DET]
    const int wave  = threadIdx.x >> 5;
    const int lane  = threadIdx.x & 31;
    const int gw    = blockIdx.x * 8 + wave;
    const int tileM = (gw & 31) << 4;             // n-tile  (32 tiles of 16)
    const int tileN = (gw >> 5) << 4;             // c-tile  (64 tiles of 16)
    const int half  = lane >> 4;                  // 0: K={0,1}, 1: K={2,3}
    const int l16   = lane & 15;

    // A-fragment source: row (tileM+l16) of H, contiguous pair of K values.
    const float* aptr = H + (tileM + l16) * DET + half * 2;

    // B-fragment source: sinogram column c = tileN + l16.
    //   S[k,c] address = (c/512)*DET*NANG + k*NANG + (c%512); lane-contiguous in c.
    const int c = tileN + l16;
    const float* bptr = sino + (size_t)(c >> 9) * (DET * NANG)
                             + (c & (NANG - 1))
                             + (size_t)(half * 2) * NANG;

    v8f acc = {};
    #pragma unroll 4
    for (int kk = 0; kk < DET; kk += 4) {
        const v2f afrag = *(const v2f*)(aptr + kk);        // H[m][kk+2h .. +1]
        v2f bfrag;
        bfrag.x = bptr[(size_t)kk * NANG];                 // S[kk+2h  ][c]
        bfrag.y = bptr[(size_t)kk * NANG + NANG];          // S[kk+2h+1][c]
        acc = __builtin_amdgcn_wmma_f32_16x16x4_f32(
                  /*neg_a=*/false, afrag,
                  /*neg_b=*/false, bfrag,
                  /*c_mod=*/(short)0, acc,
                  /*reuse_a=*/false, /*reuse_b=*/false);
    }

    // D layout: VGPR r -> (M = r + 8*half, N = l16). For a fixed lane the 8
    // accumulators are consecutive in n when stored transposed -> 2x b128.
    float* out = Ft + (size_t)c * DET + tileM + half * 8;
    v4f lo = { acc[0], acc[1], acc[2], acc[3] };
    v4f hi = { acc[4], acc[5], acc[6], acc[7] };
    *(v4f*)(out)     = lo;
    *(v4f*)(out + 4) = hi;
}

// ---------------------------------------------------------------------------
// Backprojection: out[b,i,j] = pi/(2*NANG) * sum_a lerp(Ft[b*NANG+a][:], pos)
//   pos = (j-181)*cos(a) - (i-181)*sin(a) + 256
// Adjacent j -> adjacent detector cells -> coalesced Ft reads (L2 resident).
// ---------------------------------------------------------------------------
__global__ void __launch_bounds__(256)
fbp_backproject(const float* __restrict__ Ft,
                const float* __restrict__ cosT,
                const float* __restrict__ sinT,
                float* __restrict__ out) {
    const int idx = blockIdx.x * blockDim.x + threadIdx.x;
    if (idx >= 2 * OUT * OUT) return;

    const int b = idx / (OUT * OUT);
    const int r = idx - b * (OUT * OUT);
    const int i = r / OUT;
    const int j = r - i * OUT;

    const float xg = (float)(j - OUT / 2);   // multiplies cos (ypr in reference)
    const float yg = (float)(i - OUT / 2);   // multiplies sin (xpr in reference)

    const float* Fb = Ft + (size_t)b * NANG * DET;
    float acc = 0.0f;

    for (int a = 0; a < NANG; ++a) {
        const float pos = xg * cosT[a] - yg * sinT[a] + (float)(DET / 2);
        if (pos >= 0.0f && pos <= (float)(DET - 1)) {
            const float fl = floorf(pos);
            int i0 = (int)fl;
            if (i0 < 0) i0 = 0;
            int i1 = i0 + 1; if (i1 > DET - 1) i1 = DET - 1;
            const float w = pos - fl;
            const float* row = Fb + (size_t)a * DET;
            const float v0 = row[i0];
            const float v1 = row[i1];
            acc += v0 + w * (v1 - v0);
        }
    }
    out[idx] = acc * (PI_F / (2.0f * (float)NANG));
}

// ---------------------------------------------------------------------------
// Launch. Workspace layout (needs 3 MB + 4 KB):
//   [0, 1MB)        H        512*512 f32
//   [1MB, 3MB)      Ft       1024*512 f32
//   [3MB, 3MB+2KB)  cosT     512 f32
//   [.., +2KB)      sinT     512 f32
// ---------------------------------------------------------------------------
extern "C" void kernel_launch(void* const* d_in, const int* in_sizes, int n_in,
                              void* d_out, int out_size, void* d_ws, size_t ws_size,
                              hipStream_t stream) {
    const float* sino = (const float*)d_in[0];      // (2, 512, 512, 1) f32
    float* out = (float*)d_out;                     // (2, 362, 362, 1) f32

    char* ws = (char*)d_ws;
    float* H    = (float*)(ws);
    float* Ft   = (float*)(ws + (size_t)DET * DET * sizeof(float));
    float* cosT = (float*)(ws + (size_t)DET * DET * sizeof(float)
                              + (size_t)NCOL * DET * sizeof(float));
    float* sinT = cosT + NANG;

    fbp_setup<<<(DET * DET + 255) / 256, 256, 0, stream>>>(H, cosT, sinT);

    // 32 (M) x 64 (N) tiles = 2048 waves = 256 blocks of 8 waves.
    fbp_filter_wmma<<<256, 256, 0, stream>>>(sino, H, Ft);

    const int npix = 2 * OUT * OUT;
    fbp_backproject<<<(npix + 255) / 256, 256, 0, stream>>>(Ft, cosT, sinT, out);
}